// DGG_LearnableK_Small_47536698032965
// MI455X (gfx1250) — compile-verified
//
#include <hip/hip_runtime.h>
#include <hip/hip_bf16.h>

// Problem constants (from reference)
#define B_DIM   4
#define N_DIM   2048
#define IN_DIM  128
#define LATENT  256
#define BN      (B_DIM * N_DIM)          // 8192 rows
#define INTERVAL 7.0f
#define HS_START 2.0f

typedef __attribute__((ext_vector_type(2))) float        v2f;
typedef __attribute__((ext_vector_type(4))) float        v4f;
typedef __attribute__((ext_vector_type(8))) float        v8f;
typedef __attribute__((ext_vector_type(4))) unsigned int v4u;
typedef __attribute__((ext_vector_type(4))) int          v4i;
typedef __attribute__((ext_vector_type(8))) int          v8i;

// ---------------------------------------------------------------------------
// Kernel 1: fold the last two linear layers.
//   w_eff[i] = sum_j W_mu2[i][j] * W_kp[j]
//   c0      = sum_j b_mu2[j] * W_kp[j] + b_kp
// ---------------------------------------------------------------------------
__global__ __launch_bounds__(LATENT) void prep_kernel(
    const float* __restrict__ W_mu2, const float* __restrict__ b_mu2,
    const float* __restrict__ W_kp,  const float* __restrict__ b_kp,
    float* __restrict__ w_eff, float* __restrict__ c0)
{
    const int i = threadIdx.x;             // 0..255
    float s = 0.0f;
    #pragma unroll 8
    for (int j = 0; j < LATENT; ++j)
        s += W_mu2[(size_t)i * LATENT + j] * W_kp[j];
    w_eff[i] = s;
    if (i == 0) {
        float t = b_kp[0];
        for (int j = 0; j < LATENT; ++j)
            t += b_mu2[j] * W_kp[j];
        c0[0] = t;
    }
}

// ---------------------------------------------------------------------------
// Kernel 2: shift[row] = 7 * ( relu(x[row]·W_mu1 + b_mu1) · w_eff + c0 )
// One block per 16-row tile of x (512 blocks). The 16x128 f32 x-tile is
// staged into LDS by the Tensor Data Mover (one TDM descriptor issued by
// wave 0, tracked with TENSORcnt). 16 waves; wave w computes the 16x16
// H-tile for columns [16w,16w+16) with f32 WMMA (16x16x4, K-loop over 128),
// then fuses the dot with w_eff via lane shuffles + LDS float atomics.
// ---------------------------------------------------------------------------
__global__ __launch_bounds__(512) void kshift_kernel(
    const float* __restrict__ x,
    const float* __restrict__ W_mu1, const float* __restrict__ b_mu1,
    const float* __restrict__ w_eff, const float* __restrict__ c0p,
    float* __restrict__ shift_out)
{
    __shared__ float lds_x[16 * IN_DIM];   // 8 KB x-tile
    __shared__ float acc[16];              // per-row dot accumulators

    const int tid  = threadIdx.x;
    const int mt   = blockIdx.x;           // 0..511 (row tile)
    const int wave = tid >> 5;             // 0..15  -> column tile nt
    const int lane = tid & 31;
    const int half = lane >> 4;            // 0: lanes 0-15, 1: lanes 16-31
    const int lm   = lane & 15;

    if (tid < 16) acc[tid] = 0.0f;

    // --- TDM stage: DMA the 16x128 f32 tile Global -> LDS (wave 0 only) ---
    if (wave == 0) {
        const unsigned long long gaddr =
            (unsigned long long)(uintptr_t)(x + (size_t)mt * 16 * IN_DIM);
        const unsigned int lds_base = (unsigned int)(uintptr_t)&lds_x[0];

        // D# group 0: count=1 | lds_addr | global_addr[56:0] | type=2
        v4u g0;
        g0.x = 1u;                                             // valid, 1 desc
        g0.y = lds_base;                                       // lds_addr
        g0.z = (unsigned int)(gaddr & 0xFFFFFFFFu);            // gaddr[31:0]
        g0.w = (unsigned int)((gaddr >> 32) & 0x01FFFFFFu)     // gaddr[56:32]
             | (2u << 30);                                     // type = image
        // D# group 1: mask=0, data_size=4B, dim0=128, dim1=8192,
        //             tile 128x16, dim0_stride=128
        v8i g1 = { 0x00020000,                 // data_size=2 (4 bytes)
                   (IN_DIM & 0xFFFF) << 16,    // tensor_dim0 lo16 @ [63:48]
                   (BN & 0xFFFF) << 16,        // dim0 hi16 | tensor_dim1 lo16
                   (IN_DIM & 0xFFFF) << 16,    // dim1 hi16 | tile_dim0=128
                   16,                         // tile_dim1=16 rows
                   IN_DIM,                     // tensor_dim0_stride = 128
                   0, 0 };
        v4i g2 = {};                           // 2-D tensor: groups 2/3 unused
        v4i g3 = {};
        v8i g4 = {};                           // extra group (clang-23 6-arg form)
        __builtin_amdgcn_tensor_load_to_lds(g0, g1, g2, g3, g4, /*cpol=*/0);
        __builtin_amdgcn_s_wait_tensorcnt(0);
    }
    __syncthreads();

    const int n_g = wave * 16 + lm;        // global column in [0,256)

    // A (16x4 f32): lane = {M=lm}; VGPR0/1 hold K = 2*half + {0,1}
    // B (4x16 f32): lane = {N=lm}; VGPR0/1 hold K-row = 2*half + {0,1}
    v8f c = {};
    #pragma unroll
    for (int k0 = 0; k0 < IN_DIM; k0 += 4) {
        const int ka = k0 + 2 * half;
        v2f a; a.x = lds_x[lm * IN_DIM + ka];
               a.y = lds_x[lm * IN_DIM + ka + 1];
        v2f b; b.x = W_mu1[(size_t)ka * LATENT + n_g];
               b.y = W_mu1[(size_t)(ka + 1) * LATENT + n_g];
        c = __builtin_amdgcn_wmma_f32_16x16x4_f32(
                /*neg_a=*/false, a, /*neg_b=*/false, b,
                /*c_mod=*/(short)0, c, /*reuse_a=*/false, /*reuse_b=*/false);
    }

    // bias + relu, then dot with w_eff; reduce over N (lanes of a 16-group).
    const float w   = w_eff[n_g];
    const float bia = b_mu1[n_g];
    #pragma unroll
    for (int r = 0; r < 8; ++r) {
        float h = c[r] + bia;
        h = h > 0.0f ? h : 0.0f;
        float p = h * w;
        #pragma unroll
        for (int off = 1; off < 16; off <<= 1)
            p += __shfl_xor(p, off, 32);   // reduces within each 16-lane half
        if (lm == 0) {
            const int row = r + 8 * half;  // C/D layout: M = r + 8*(lane/16)
            atomicAdd(&acc[row], p);       // ds_add_f32 across the 16 waves
        }
    }
    __syncthreads();

    if (tid < 16)
        shift_out[mt * 16 + tid] = INTERVAL * (acc[tid] + c0p[0]);
}

// ---------------------------------------------------------------------------
// Kernel 3: streaming output (the bandwidth-bound part: 134 MB of stores).
//   adj[b,i,j]  = sigmoid(2 - 7j + shift[b,i])
//   idxs[b,i,j] = j   (identity permutation; argsort of a uniform row)
// One float4 + one int4 per thread, written non-temporally (write-once data,
// never re-read on device -> keep it out of L2). Sigmoid uses v_exp_f32 +
// v_rcp_f32 to stay well under the 23.3 TB/s store roof.
// ---------------------------------------------------------------------------
__device__ __forceinline__ float fast_sigmoid(float t) {
    return __builtin_amdgcn_rcpf(1.0f + __expf(-t));
}

__global__ __launch_bounds__(256) void fill_kernel(
    const float* __restrict__ shift,
    float* __restrict__ adj, int* __restrict__ idxs)
{
    const int gid = blockIdx.x * 256 + threadIdx.x;     // < BN * N/4 = 2^22
    const int gpr = N_DIM >> 2;                         // 512 groups per row
    const int row = gid / gpr;                          // wave-uniform
    const int j0  = (gid - row * gpr) << 2;

    const float s = shift[row] + HS_START - INTERVAL * (float)j0;

    v4f v;
    v.x = fast_sigmoid(s);
    v.y = fast_sigmoid(s - INTERVAL);
    v.z = fast_sigmoid(s - 2.0f * INTERVAL);
    v.w = fast_sigmoid(s - 3.0f * INTERVAL);
    __builtin_nontemporal_store(v, reinterpret_cast<v4f*>(adj) + gid);

    v4i iv = { j0, j0 + 1, j0 + 2, j0 + 3 };
    __builtin_nontemporal_store(iv, reinterpret_cast<v4i*>(idxs) + gid);
}

// ---------------------------------------------------------------------------
extern "C" void kernel_launch(void* const* d_in, const int* in_sizes, int n_in,
                              void* d_out, int out_size, void* d_ws, size_t ws_size,
                              hipStream_t stream)
{
    // setup_inputs() order:
    // 0:x 1:temp 2:W_in 3:b_in 4:w_dist 5:b_dist
    // 6:W_mu1 7:b_mu1 8:W_mu2 9:b_mu2 10:W_kp 11:b_kp
    const float* x     = (const float*)d_in[0];
    const float* W_mu1 = (const float*)d_in[6];
    const float* b_mu1 = (const float*)d_in[7];
    const float* W_mu2 = (const float*)d_in[8];
    const float* b_mu2 = (const float*)d_in[9];
    const float* W_kp  = (const float*)d_in[10];
    const float* b_kp  = (const float*)d_in[11];

    // Workspace layout (floats): [0,256) w_eff | [256] c0 | [512, 512+8192) shift
    float* ws    = (float*)d_ws;
    float* w_eff = ws;
    float* c0    = ws + 256;
    float* shift = ws + 512;

    // Output: adj_hard [B,N,N] f32, then idxs [B,N,N] int32 (concatenated).
    const size_t tile_elems = (size_t)B_DIM * N_DIM * N_DIM;  // 16,777,216
    float* adj  = (float*)d_out;
    int*   idxs = (int*)(adj + tile_elems);

    prep_kernel<<<1, LATENT, 0, stream>>>(W_mu2, b_mu2, W_kp, b_kp, w_eff, c0);
    kshift_kernel<<<BN / 16, 512, 0, stream>>>(x, W_mu1, b_mu1, w_eff, c0, shift);
    fill_kernel<<<(int)(tile_elems / 4 / 256), 256, 0, stream>>>(shift, adj, idxs);
}